// HLSQoREstimator_74706661147046
// MI455X (gfx1250) — compile-verified
//
#include <hip/hip_runtime.h>
#include <hip/hip_bf16.h>

// ---------------------------------------------------------------------------
// MI455X (gfx1250) implementation of the heterogeneous-GNN QoR estimator.
//   - Edge aggregation: global_atomic_add_f32 scatter (L2-resident targets)
//   - All H=64 linears: V_WMMA_F32_16X16X4_F32 (fp32 matrix path, wave32),
//     weights staged in LDS by the Tensor Data Mover (tensor_load_to_lds)
//   - Head: scalar kernels (B=64, negligible)
// ---------------------------------------------------------------------------

typedef __attribute__((ext_vector_type(2))) float v2f;
typedef __attribute__((ext_vector_type(8))) float v8f;
typedef __attribute__((ext_vector_type(4))) unsigned int u32x4;
typedef __attribute__((ext_vector_type(8))) int i32x8;
typedef __attribute__((ext_vector_type(4))) int i32x4;

#define NOP 400000
#define NVAR 300000
#define NBLK 20000
#define BATCH 64

// ---------------------------------------------------------------------------
// TDM load of a 64x64 f32 tile (16 KB) from global into LDS.
// D# per CDNA5 ISA ch.8: group0 = {count=1, lds_addr, global_addr, type=2};
// group1 = {mask=0, data_size=4B, dims/tile 64x64, stride0=64}; groups 2/3 = 0
// (2-D tensor). Issued by one wave; completion via s_wait_tensorcnt.
// ---------------------------------------------------------------------------
__device__ inline void tdm_load_w64(const float* W, float* ldsW)
{
  const unsigned long long ga = (unsigned long long)(uintptr_t)W;
  // LDS aperture maps flat addr[31:0] directly to the LDS byte offset.
  const unsigned lds_base = (unsigned)(unsigned long long)(uintptr_t)(void*)ldsW;

  u32x4 g0;
  g0[0] = 1u;                                            // count=1 (valid user D#)
  g0[1] = lds_base;                                      // lds_addr (bytes)
  g0[2] = (unsigned)(ga & 0xFFFFFFFFull);                // global_addr[31:0]
  g0[3] = (unsigned)((ga >> 32) & 0x1FFFFFFull)          // global_addr[56:32]
        | (2u << 30);                                    // type=2 ("image")

  i32x8 g1;
  g1[0] = (int)(2u << 16);    // workgroup_mask=0, data_size=2 (4 bytes), no flags
  g1[1] = (int)(64u << 16);   // atomic_barrier_addr=0 | tensor_dim0[15:0]=64
  g1[2] = (int)(64u << 16);   // tensor_dim0[31:16]=0  | tensor_dim1[15:0]=64
  g1[3] = (int)(64u << 16);   // tensor_dim1[31:16]=0  | tile_dim0=64
  g1[4] = 64;                 // tile_dim1=64 | tile_dim2=0
  g1[5] = 64;                 // tensor_dim0_stride[31:0]=64 (data_size units)
  g1[6] = 0;                  // stride0[47:32]=0 | tensor_dim1_stride[15:0]=0
  g1[7] = 0;

  i32x4 g2 = {};              // 2-D tensor: groups 2/3 unused
  i32x4 g3 = {};

#if defined(__clang_major__) && (__clang_major__ >= 23)
  i32x8 g4 = {};
  __builtin_amdgcn_tensor_load_to_lds(g0, g1, g2, g3, g4, 0);
#else
  __builtin_amdgcn_tensor_load_to_lds(g0, g1, g2, g3, 0);
#endif
  __builtin_amdgcn_s_wait_tensorcnt(0);
}

// ---------------------------------------------------------------------------
// Fused 64x64 GEMM on the WMMA f32 path.
// Y[r,c] = act( sum_k (X[r,k]*scale_r) * W[k,c] + bias[c] + ADD[r,c] ) + RES[r,c]
//   scale_r = 1/max(cnt[r],1) when cnt != nullptr (mean aggregation)
//   act = ReLU when do_relu, identity otherwise (residual added post-act)
// Block = 256 threads = 8 waves; each wave owns a 16x64 output tile.
// A fragment (16x4 f32): lanes 0-15 hold K={kb,kb+1}, lanes 16-31 K={kb+2,kb+3}.
// B fragment (4x16 f32): VGPR0 = rows K{0,2}, VGPR1 = rows K{1,3} (half-split).
// D (16x16 f32, 8 VGPRs): VGPR r -> M=r (lanes 0-15) / M=r+8 (lanes 16-31).
// ---------------------------------------------------------------------------
__global__ __launch_bounds__(256) void gemm64_wmma(
    const float* __restrict__ X, const float* __restrict__ W,
    const float* __restrict__ bias, const float* __restrict__ ADD,
    const float* __restrict__ RES, const float* __restrict__ cnt,
    float* __restrict__ Y, int Nrows, int do_relu)
{
  __shared__ float ldsW[64 * 64];
  if ((threadIdx.x >> 5) == 0) {           // one wave issues the TDM DMA
    tdm_load_w64(W, ldsW);
  }
  __syncthreads();                          // LDS tile visible to all waves

  const int wave = threadIdx.x >> 5;
  const int lane = threadIdx.x & 31;
  const int half = lane >> 4;      // 0: lanes 0-15, 1: lanes 16-31
  const int ml   = lane & 15;
  const int row0 = (blockIdx.x * 8 + wave) * 16;
  if (row0 >= Nrows) return;       // whole-tile guard (all N are 16-aligned)

  const int arow = row0 + ml;      // A rows M=0..15 live in ml for both halves
  float scale = 1.0f;
  if (cnt) scale = 1.0f / fmaxf(cnt[arow], 1.0f);

  v8f acc[4] = {};                 // four 16x16 N-tiles -> 64 output columns
  const float* xrow = X + (size_t)arow * 64;

#pragma unroll
  for (int kb = 0; kb < 64; kb += 4) {
    const int k0 = kb + 2 * half;
    v2f a;
    a.x = xrow[k0] * scale;
    a.y = xrow[k0 + 1] * scale;
    const float* w0 = &ldsW[k0 * 64];
    const float* w1 = &ldsW[(k0 + 1) * 64];
#pragma unroll
    for (int nt = 0; nt < 4; ++nt) {
      v2f b;
      b.x = w0[nt * 16 + ml];
      b.y = w1[nt * 16 + ml];
      acc[nt] = __builtin_amdgcn_wmma_f32_16x16x4_f32(
          false, a, false, b, (short)0, acc[nt], false, false);
    }
  }

#pragma unroll
  for (int nt = 0; nt < 4; ++nt) {
    const int col = nt * 16 + ml;
    const float bv = bias[col];
#pragma unroll
    for (int r = 0; r < 8; ++r) {
      const int rowg = row0 + half * 8 + r;
      const size_t o = (size_t)rowg * 64 + col;
      float v = acc[nt][r] + bv;
      if (ADD) v += ADD[o];
      if (do_relu) v = fmaxf(v, 0.0f);
      if (RES) v += RES[o];        // residual is added after activation
      Y[o] = v;
    }
  }
}

// ---------------------------------------------------------------------------
// Edge scatter: S[dst,:] += X[src,:], cnt[dst] += 1. One thread per (edge,feat).
// ---------------------------------------------------------------------------
__global__ __launch_bounds__(256) void edge_scatter(
    const float* __restrict__ Xsrc, const int* __restrict__ ei, int E,
    float* __restrict__ S, float* __restrict__ cnt)
{
  const long long gid = (long long)blockIdx.x * 256 + threadIdx.x;
  if (gid >= (long long)E * 64) return;
  const int e = (int)(gid >> 6);
  const int f = (int)(gid & 63);
  const int src = ei[e];
  const int dst = ei[E + e];
  atomicAdd(&S[(size_t)dst * 64 + f], Xsrc[(size_t)src * 64 + f]);
  if (f == 0) atomicAdd(&cnt[dst], 1.0f);
}

__device__ inline void atomicMaxF(float* addr, float val)
{
  int* ia = (int*)addr;
  int old = *ia;
  while (__int_as_float(old) < val) {
    const int assumed = old;
    old = atomicCAS(ia, assumed, __float_as_int(val));
    if (old == assumed) break;
  }
}

__global__ __launch_bounds__(256) void pool_scatter(
    const float* __restrict__ X, const int* __restrict__ batch, int N,
    float* __restrict__ sum, float* __restrict__ cnt, float* __restrict__ mx)
{
  const long long gid = (long long)blockIdx.x * 256 + threadIdx.x;
  if (gid >= (long long)N * 64) return;
  const int n = (int)(gid >> 6);
  const int f = (int)(gid & 63);
  const int b = batch[n];
  const float v = X[(size_t)n * 64 + f];
  atomicAdd(&sum[b * 64 + f], v);
  atomicMaxF(&mx[b * 64 + f], v);
  if (f == 0) atomicAdd(&cnt[b], 1.0f);
}

// P[b, 0:64]=mean, P[b, 64:128]=sum, P[b, 128:192]=max   (din=192 for aggr lin)
__global__ void pool_finalize(const float* __restrict__ sum,
                              const float* __restrict__ cnt,
                              const float* __restrict__ mx,
                              float* __restrict__ P)
{
  const int b = blockIdx.x;
  const int f = threadIdx.x;
  const float s = sum[b * 64 + f];
  const float c = fmaxf(cnt[b], 1.0f);
  P[b * 192 + f] = s / c;
  P[b * 192 + 64 + f] = s;
  P[b * 192 + 128 + f] = mx[b * 64 + f];
}

__global__ void fill_kernel(float* __restrict__ p, float v, int n)
{
  const int i = blockIdx.x * 256 + threadIdx.x;
  if (i < n) p[i] = v;
}

// Generic small dense: Y[r*ostride + ooff + c] = X[r,:] @ W[:,c] + bias[c]
__global__ void dense_kernel(const float* __restrict__ X, const float* __restrict__ W,
                             const float* __restrict__ bias, float* __restrict__ Y,
                             int rows, int din, int dout, int ostride, int ooff)
{
  const int idx = blockIdx.x * 256 + threadIdx.x;
  if (idx >= rows * dout) return;
  const int r = idx / dout;
  const int c = idx % dout;
  float acc = bias[c];
  const float* xr = X + (size_t)r * din;
  for (int k = 0; k < din; ++k) acc += xr[k] * W[k * dout + c];
  Y[(size_t)r * ostride + ooff + c] = acc;
}

// In-place LayerNorm + PReLU over rows of Y[rows, d]; one block per row.
__global__ __launch_bounds__(256) void ln_prelu_kernel(
    float* __restrict__ Y, const float* __restrict__ g,
    const float* __restrict__ bta, const float* __restrict__ a_ptr, int d)
{
  __shared__ float red[256];
  const int r = blockIdx.x;
  const int t = threadIdx.x;
  float* yr = Y + (size_t)r * d;

  float s = 0.0f;
  for (int i = t; i < d; i += 256) s += yr[i];
  red[t] = s; __syncthreads();
  for (int w = 128; w > 0; w >>= 1) { if (t < w) red[t] += red[t + w]; __syncthreads(); }
  const float mean = red[0] / d;
  __syncthreads();

  float v = 0.0f;
  for (int i = t; i < d; i += 256) { const float x = yr[i] - mean; v += x * x; }
  red[t] = v; __syncthreads();
  for (int w = 128; w > 0; w >>= 1) { if (t < w) red[t] += red[t + w]; __syncthreads(); }
  const float rstd = rsqrtf(red[0] / d + 1e-5f);
  const float a = *a_ptr;

  for (int i = t; i < d; i += 256) {
    const float x = (yr[i] - mean) * rstd * g[i] + bta[i];
    yr[i] = x >= 0.0f ? x : a * x;
  }
}

// y_base head: 1 -> 16 (PReLU) -> 16, written into XA[:, 192:208]
__global__ void yb_kernel(const float* __restrict__ y_base, const float* __restrict__ a1p,
                          const float* __restrict__ b1, const float* __restrict__ w1,
                          const float* __restrict__ b2, const float* __restrict__ w2,
                          float* __restrict__ XA)
{
  const int r = threadIdx.x;
  if (r >= BATCH) return;
  const float y = y_base[r];
  const float a1 = *a1p;
  float h[16];
#pragma unroll
  for (int j = 0; j < 16; ++j) {
    const float t = y * w1[j] + b1[j];
    h[j] = t >= 0.0f ? t : a1 * t;
  }
#pragma unroll
  for (int j = 0; j < 16; ++j) {
    float acc = b2[j];
#pragma unroll
    for (int k = 0; k < 16; ++k) acc += h[k] * w2[k * 16 + j];
    XA[r * 208 + 192 + j] = acc;
  }
}

// ---------------------------------------------------------------------------
// Host orchestration. Input order: 13 arrays + params pytree leaves (jax
// flatten order: dicts by sorted key; lin -> {b, w}).
// ---------------------------------------------------------------------------
extern "C" void kernel_launch(void* const* d_in, const int* in_sizes, int n_in,
                              void* d_out, int out_size, void* d_ws, size_t ws_size,
                              hipStream_t stream)
{
  (void)in_sizes; (void)n_in; (void)out_size; (void)ws_size;
  const float* x_op    = (const float*)d_in[0];
  const float* x_var   = (const float*)d_in[1];
  const float* x_block = (const float*)d_in[2];
  const int* ei_op_op  = (const int*)d_in[3];
  const int* ei_var_op = (const int*)d_in[4];
  const int* ei_op_var = (const int*)d_in[5];
  const int* ei_blk_op = (const int*)d_in[6];
  const int* ei_op_blk = (const int*)d_in[7];
  const int* ei_blk_blk= (const int*)d_in[8];
  const int* batch_op  = (const int*)d_in[9];
  const int* batch_var = (const int*)d_in[10];
  const int* batch_blk = (const int*)d_in[11];
  const float* y_base  = (const float*)d_in[12];
  auto F = [&](int i) { return (const float*)d_in[i]; };

  // params leaf indices (sorted-key pytree flatten, starting at 13):
  // 13..18 aggr{block,op,var}{b,w}
  // 19..32 gnn[0]: neigh{op_flows_op, op_within_block, op_writes_var,
  //                      var_used_by_op}{b,w}, self{block,op,var}{b,w}
  // 33..50 gnn[1]: neigh{block_contains_op, block_next_block, op_flows_op,
  //                      op_within_block, op_writes_var, var_used_by_op}{b,w},
  //                self{block,op,var}{b,w};   51..68 gnn[2] same
  // 69..83 mlp[0..2]{a, lin.b, lin.w, ln_b, ln_g}
  // 84..85 mlp_out{b,w}; 86..89 proj{op,var}{b,w}; 90..94 yb{a1, l1.b,l1.w, l2.b,l2.w}
  const int nb_op_op[3]  = {19, 37, 55}, nw_op_op[3]  = {20, 38, 56};
  const int nb_op_blk[3] = {21, 39, 57}, nw_op_blk[3] = {22, 40, 58};
  const int nb_op_var[3] = {23, 41, 59}, nw_op_var[3] = {24, 42, 60};
  const int nb_var_op[3] = {25, 43, 61}, nw_var_op[3] = {26, 44, 62};
  const int nb_blk_op[3] = {-1, 33, 51}, nw_blk_op[3] = {-1, 34, 52};
  const int nb_blk_blk[3]= {-1, 35, 53}, nw_blk_blk[3]= {-1, 36, 54};
  const int sb_blk[3] = {27, 45, 63}, sw_blk[3] = {28, 46, 64};
  const int sb_op[3]  = {29, 47, 65}, sw_op[3]  = {30, 48, 66};
  const int sb_var[3] = {31, 49, 67}, sw_var[3] = {32, 50, 68};

  // workspace layout (fp32)
  float* ws = (float*)d_ws;
  size_t off = 0;
  auto alloc = [&](size_t n) { float* p = ws + off; off += n; return p; };
  float* Aop  = alloc((size_t)NOP * 64);  float* Bop  = alloc((size_t)NOP * 64);
  float* Gop  = alloc((size_t)NOP * 64);
  float* Avar = alloc((size_t)NVAR * 64); float* Bvar = alloc((size_t)NVAR * 64);
  float* Gvar = alloc((size_t)NVAR * 64);
  float* Ablk = alloc((size_t)NBLK * 64); float* Bblk = alloc((size_t)NBLK * 64);
  float* Gblk = alloc((size_t)NBLK * 64);
  float* S    = alloc((size_t)NOP * 64);  // per-edge-type sum (reused; max N_dst)
  float* CNT  = alloc((size_t)NOP);       // per-edge-type dst degree (reused)
  float* psum = alloc(3 * BATCH * 64);
  float* pcnt = alloc(3 * BATCH);
  float* pmax = alloc(3 * BATCH * 64);
  float* P    = alloc(3 * BATCH * 192);
  float* XA   = alloc((size_t)BATCH * 208);
  float* H1   = alloc((size_t)BATCH * 512);
  float* H2   = alloc((size_t)BATCH * 256);
  float* H3   = alloc((size_t)BATCH * 128);

  auto run_gemm = [&](const float* X, const float* W, const float* bvec,
                      const float* ADD, const float* RES, const float* cntp,
                      float* Yv, int Nrows, int relu) {
    const int blocks = (Nrows + 127) / 128;
    hipLaunchKernelGGL(gemm64_wmma, dim3(blocks), dim3(256), 0, stream,
                       X, W, bvec, ADD, RES, cntp, Yv, Nrows, relu);
  };
  auto run_edge = [&](const float* Xsrc, const int* ei, int E, int Ndst,
                      float* AGG, const float* w, const float* bvec) {
    hipMemsetAsync(S, 0, (size_t)Ndst * 64 * sizeof(float), stream);
    hipMemsetAsync(CNT, 0, (size_t)Ndst * sizeof(float), stream);
    const long long tot = (long long)E * 64;
    hipLaunchKernelGGL(edge_scatter, dim3((unsigned)((tot + 255) / 256)),
                       dim3(256), 0, stream, Xsrc, ei, E, S, CNT);
    run_gemm(S, w, bvec, /*ADD=*/AGG, nullptr, /*cnt=*/CNT, AGG, Ndst, 0);
  };

  // ---- input projection: op & var get H->H linear, block passes through ----
  run_gemm(x_op,  F(87), F(86), nullptr, nullptr, nullptr, Aop,  NOP,  0);
  run_gemm(x_var, F(89), F(88), nullptr, nullptr, nullptr, Avar, NVAR, 0);
  hipMemcpyAsync(Ablk, x_block, (size_t)NBLK * 64 * sizeof(float),
                 hipMemcpyDeviceToDevice, stream);

  float *cop = Aop, *cvar = Avar, *cblk = Ablk;
  float *top = Bop, *tvar = Bvar, *tblk = Bblk;

  for (int L = 0; L < 3; ++L) {
    hipMemsetAsync(Gop,  0, (size_t)NOP  * 64 * sizeof(float), stream);
    hipMemsetAsync(Gvar, 0, (size_t)NVAR * 64 * sizeof(float), stream);
    hipMemsetAsync(Gblk, 0, (size_t)NBLK * 64 * sizeof(float), stream);

    run_edge(cop,  ei_op_op,  2000000, NOP,  Gop,  F(nw_op_op[L]),  F(nb_op_op[L]));
    run_edge(cvar, ei_var_op, 1000000, NOP,  Gop,  F(nw_var_op[L]), F(nb_var_op[L]));
    run_edge(cop,  ei_op_var, 1000000, NVAR, Gvar, F(nw_op_var[L]), F(nb_op_var[L]));
    run_edge(cop,  ei_op_blk,  400000, NBLK, Gblk, F(nw_op_blk[L]), F(nb_op_blk[L]));
    if (L > 0) {  // block-sourced edge types excluded from layer 0
      run_edge(cblk, ei_blk_op,  400000, NOP,  Gop,  F(nw_blk_op[L]),  F(nb_blk_op[L]));
      run_edge(cblk, ei_blk_blk,  60000, NBLK, Gblk, F(nw_blk_blk[L]), F(nb_blk_blk[L]));
    }

    const int relu = (L < 2) ? 1 : 0;
    run_gemm(cop,  F(sw_op[L]),  F(sb_op[L]),  Gop,  cop,  nullptr, top,  NOP,  relu);
    run_gemm(cvar, F(sw_var[L]), F(sb_var[L]), Gvar, cvar, nullptr, tvar, NVAR, relu);
    const float* blkres = (L == 0) ? nullptr : cblk;  // no block residual at L0
    run_gemm(cblk, F(sw_blk[L]), F(sb_blk[L]), Gblk, blkres, nullptr, tblk, NBLK, relu);

    float* t;
    t = cop;  cop  = top;  top  = t;
    t = cvar; cvar = tvar; tvar = t;
    t = cblk; cblk = tblk; tblk = t;
  }

  // ---- graph pooling (mean/sum/max) + aggr linear, NT order: block, op, var
  auto run_pool = [&](const float* Xf, const int* batch, int Nn, int slot,
                      const float* aw, const float* ab, int colOff) {
    float* sb = psum + slot * BATCH * 64;
    float* cb = pcnt + slot * BATCH;
    float* mb = pmax + slot * BATCH * 64;
    float* Pb = P    + slot * BATCH * 192;
    hipMemsetAsync(sb, 0, BATCH * 64 * sizeof(float), stream);
    hipMemsetAsync(cb, 0, BATCH * sizeof(float), stream);
    hipLaunchKernelGGL(fill_kernel, dim3((BATCH * 64 + 255) / 256), dim3(256), 0,
                       stream, mb, -3.402823466e38f, BATCH * 64);
    const long long tot = (long long)Nn * 64;
    hipLaunchKernelGGL(pool_scatter, dim3((unsigned)((tot + 255) / 256)),
                       dim3(256), 0, stream, Xf, batch, Nn, sb, cb, mb);
    hipLaunchKernelGGL(pool_finalize, dim3(BATCH), dim3(64), 0, stream, sb, cb, mb, Pb);
    hipLaunchKernelGGL(dense_kernel, dim3((BATCH * 64 + 255) / 256), dim3(256), 0,
                       stream, Pb, aw, ab, XA, BATCH, 192, 64, 208, colOff);
  };
  run_pool(cblk, batch_blk, NBLK, 0, F(14), F(13), 0);
  run_pool(cop,  batch_op,  NOP,  1, F(16), F(15), 64);
  run_pool(cvar, batch_var, NVAR, 2, F(18), F(17), 128);

  // y_base branch -> XA[:, 192:208]
  hipLaunchKernelGGL(yb_kernel, dim3(1), dim3(64), 0, stream,
                     y_base, F(90), F(91), F(92), F(93), F(94), XA);

  // ---- MLP head: 208 -> 512 -> 256 -> 128 -> 1, LN + PReLU between ----
  hipLaunchKernelGGL(dense_kernel, dim3((BATCH * 512 + 255) / 256), dim3(256), 0,
                     stream, XA, F(71), F(70), H1, BATCH, 208, 512, 512, 0);
  hipLaunchKernelGGL(ln_prelu_kernel, dim3(BATCH), dim3(256), 0, stream,
                     H1, F(73), F(72), F(69), 512);
  hipLaunchKernelGGL(dense_kernel, dim3((BATCH * 256 + 255) / 256), dim3(256), 0,
                     stream, H1, F(76), F(75), H2, BATCH, 512, 256, 256, 0);
  hipLaunchKernelGGL(ln_prelu_kernel, dim3(BATCH), dim3(256), 0, stream,
                     H2, F(78), F(77), F(74), 256);
  hipLaunchKernelGGL(dense_kernel, dim3((BATCH * 128 + 255) / 256), dim3(256), 0,
                     stream, H2, F(81), F(80), H3, BATCH, 256, 128, 128, 0);
  hipLaunchKernelGGL(ln_prelu_kernel, dim3(BATCH), dim3(256), 0, stream,
                     H3, F(83), F(82), F(79), 128);
  hipLaunchKernelGGL(dense_kernel, dim3((BATCH * 1 + 255) / 256), dim3(256), 0,
                     stream, H3, F(85), F(84), (float*)d_out, BATCH, 128, 1, 1, 0);
}